// BillehColumn_54142357733913
// MI455X (gfx1250) — compile-verified
//
#include <hip/hip_runtime.h>
#include <hip/hip_bf16.h>

// Problem constants (from reference)
#define NN   50000
#define RR   10
#define DD   5
#define EE   10000000
#define RN   (RR * NN)      // 500000
#define DN   (DD * NN)      // 250000

// d_out layout (floats), in reference return order
#define OFF_Z     0
#define OFF_OUTV  (NN)            // 50000
#define OFF_ZBUF  (2 * NN)        // 100000
#define OFF_NEWV  (OFF_ZBUF + DN) // 350000
#define OFF_R     (OFF_NEWV + NN) // 400000
#define OFF_A1    (OFF_R + NN)    // 450000
#define OFF_A2    (OFF_A1 + NN)   // 500000
#define OFF_PR    (OFF_A2 + NN)   // 550000
#define OFF_PSC   (OFF_PR + RN)   // 1050000

__global__ __launch_bounds__(256) void billeh_zero_kernel(float* __restrict__ p, int n) {
    int i = blockIdx.x * blockDim.x + threadIdx.x;
    if (i < n) p[i] = 0.0f;
}

// Sparse scatter: i_rec[row] += w * z[col], skipping the ~98% of edges where z==0.
// 2 edges per thread: int4 (b128) index load + float2 (b64) weight load, fully coalesced.
__global__ __launch_bounds__(256) void billeh_edge_kernel(
    const float* __restrict__ w, const int4* __restrict__ idx2,
    const float* __restrict__ z, float* __restrict__ acc) {
    int t = blockIdx.x * blockDim.x + threadIdx.x;
    if (t * 2 >= EE) return;
    int4   p  = idx2[t];                      // edge 2t: (x=row,y=col); edge 2t+1: (z=row,w=col)
    float2 wv = ((const float2*)w)[t];
    float z0 = z[p.y];
    float z1 = z[p.w];
    if (z0 != 0.0f) atomicAdd(&acc[p.x], wv.x * z0);
    if (z1 != 0.0f) atomicAdd(&acc[p.z], wv.y * z1);
}

// Per-receptor update (R*N lanes, fully coalesced). LR_SCALE=1, DT=1.
__global__ __launch_bounds__(256) void billeh_receptor_kernel(
    const float* __restrict__ acc, const float* __restrict__ inputs,
    const float* __restrict__ psc_rise, const float* __restrict__ psc,
    const float* __restrict__ syn_decay, const float* __restrict__ psc_init,
    float* __restrict__ out_psc_rise, float* __restrict__ out_psc) {
    int j = blockIdx.x * blockDim.x + threadIdx.x;
    if (j >= RN) return;
    float sd     = syn_decay[j];
    float pr     = psc_rise[j];
    float rec_in = acc[j] + inputs[j];                    // i_rec*LR_SCALE + inputs
    out_psc_rise[j] = fmaf(sd, pr, rec_in * psc_init[j]); // syn_decay*psc_rise + rec_in*psc_initial
    out_psc[j]      = fmaf(psc[j], sd, sd * pr);          // psc*syn_decay + DT*syn_decay*psc_rise
}

// Per-neuron update. Stages the block's psc slab (256 neurons x 10 receptors = 10KB)
// into LDS via the CDNA5 async-to-LDS engine, then reduces from LDS (stride-10 over
// 64 banks -> conflict-free), avoiding stride-40B vector loads from global.
__global__ __launch_bounds__(256) void billeh_neuron_kernel(
    const float* __restrict__ psc,   const float* __restrict__ z_buf,
    const float* __restrict__ v,     const float* __restrict__ r,
    const float* __restrict__ asc1,  const float* __restrict__ asc2,
    const float* __restrict__ v_th,  const float* __restrict__ e_l,
    const float* __restrict__ v_reset, const float* __restrict__ g,
    const float* __restrict__ decay, const float* __restrict__ cf,
    const float* __restrict__ t_ref, const float* __restrict__ k,
    const float* __restrict__ amps,  const float* __restrict__ vscale,
    const float* __restrict__ voffset, float* __restrict__ out) {
    __shared__ float spsc[256 * RR];
    const int tid = threadIdx.x;
    const int n0  = blockIdx.x * 256;

    // Async stage psc[n0*RR .. n0*RR+2560) -> LDS (coalesced b32, 10 in-flight ops/wave)
#pragma unroll
    for (int it = 0; it < RR; ++it) {
        int gi = n0 * RR + it * 256 + tid;
        if (gi < RN) {
            unsigned loff = (unsigned)(size_t)(&spsc[it * 256 + tid]); // low 32 bits = LDS byte offset
            unsigned long long ga = (unsigned long long)(const void*)(psc + gi);
            asm volatile("global_load_async_to_lds_b32 %0, %1, off"
                         :: "v"(loff), "v"(ga) : "memory");
        }
    }
    asm volatile("s_wait_asynccnt 0" ::: "memory");
    __syncthreads();

    const int n = n0 + tid;
    if (n >= NN) return;

    float s = 0.0f;
#pragma unroll
    for (int j = 0; j < RR; ++j) s += spsc[tid * RR + j];  // input_current from OLD psc

    float prev_z = z_buf[n];                               // delay slot 0
    float rn = fmaxf(r[n] + prev_z * t_ref[n] - 1.0f, 0.0f); // DT=1
    float kk0 = 1.0f / (1.0f + __expf(-k[2 * n]));
    float kk1 = 1.0f / (1.0f + __expf(-k[2 * n + 1]));
    // use expf for the state decay factors (accuracy-sensitive)
    float a1 = asc1[n], a2 = asc2[n];
    float na1 = expf(-kk0) * a1 + prev_z * amps[2 * n];
    float na2 = expf(-kk1) * a2 + prev_z * amps[2 * n + 1];
    float el  = e_l[n];
    float nv  = decay[n] * v[n] + cf[n] * (s + a1 + a2 + g[n] * el); // OLD asc
    float vth = v_th[n];
    float vsc = (nv - vth) / (vth - el);
    float z = (vsc > 0.0f) ? 1.0f : 0.0f;
    if (rn > 0.0f) z = 0.0f;
    if (z > 0.5f)  nv = v_reset[n];

    out[OFF_Z    + n] = z;
    out[OFF_OUTV + n] = fmaf(nv, vscale[n], voffset[n]);
    out[OFF_NEWV + n] = nv;
    out[OFF_R    + n] = rn;
    out[OFF_A1   + n] = na1;
    out[OFF_A2   + n] = na2;
    // delay-buffer shift: slot 0 = new_z, slot d = old slot d-1
    out[OFF_ZBUF + n] = z;
#pragma unroll
    for (int d = 0; d < DD - 1; ++d)
        out[OFF_ZBUF + (d + 1) * NN + n] = z_buf[d * NN + n];
}

extern "C" void kernel_launch(void* const* d_in, const int* in_sizes, int n_in,
                              void* d_out, int out_size, void* d_ws, size_t ws_size,
                              hipStream_t stream) {
    (void)in_sizes; (void)n_in; (void)out_size; (void)ws_size;
    const float* inputs    = (const float*)d_in[0];
    const float* z_buf     = (const float*)d_in[1];
    const float* v         = (const float*)d_in[2];
    const float* r         = (const float*)d_in[3];
    const float* asc_1     = (const float*)d_in[4];
    const float* asc_2     = (const float*)d_in[5];
    const float* psc_rise  = (const float*)d_in[6];
    const float* psc       = (const float*)d_in[7];
    const float* rec_w     = (const float*)d_in[8];
    const int*   rec_idx   = (const int*)d_in[9];
    const float* v_th      = (const float*)d_in[10];
    const float* e_l       = (const float*)d_in[11];
    const float* v_reset   = (const float*)d_in[12];
    const float* g         = (const float*)d_in[13];
    const float* decay     = (const float*)d_in[14];
    const float* cf        = (const float*)d_in[15];
    const float* t_ref     = (const float*)d_in[16];
    const float* k         = (const float*)d_in[17];
    const float* amps      = (const float*)d_in[18];
    const float* syn_decay = (const float*)d_in[19];
    const float* psc_init  = (const float*)d_in[20];
    const float* vscale    = (const float*)d_in[21];
    const float* voffset   = (const float*)d_in[22];

    float* out = (float*)d_out;
    float* acc = (float*)d_ws;   // R*N accumulator (2 MB)

    billeh_zero_kernel<<<(RN + 255) / 256, 256, 0, stream>>>(acc, RN);
    billeh_edge_kernel<<<(EE / 2 + 255) / 256, 256, 0, stream>>>(
        rec_w, (const int4*)rec_idx, z_buf, acc);
    billeh_receptor_kernel<<<(RN + 255) / 256, 256, 0, stream>>>(
        acc, inputs, psc_rise, psc, syn_decay, psc_init,
        out + OFF_PR, out + OFF_PSC);
    billeh_neuron_kernel<<<(NN + 255) / 256, 256, 0, stream>>>(
        psc, z_buf, v, r, asc_1, asc_2, v_th, e_l, v_reset, g, decay, cf,
        t_ref, k, amps, vscale, voffset, out);
}